// DisentangleLossBatch_68023692034358
// MI455X (gfx1250) — compile-verified
//
#include <hip/hip_runtime.h>
#include <hip/hip_bf16.h>
#include <math.h>

// Problem constants (match reference shapes)
#define BB 32
#define NN 1024
#define DD 512      // codebook rows / top-k domain
#define EE 256      // codebook row length
#define BN (BB * NN)
#define GN 8        // GROUP_NUM

typedef __attribute__((ext_vector_type(2))) float v2f;
typedef __attribute__((ext_vector_type(8))) float v8f;

// ---------------------------------------------------------------------------
// Zero the 64-float pair accumulator
// ---------------------------------------------------------------------------
__global__ void dl_zero_acc(float* __restrict__ acc64) {
    acc64[threadIdx.x] = 0.0f;
}

// ---------------------------------------------------------------------------
// Per-row inverse L2 norm of codebook: inv[d] = 1 / max(||cb[d,:]||, 1e-12)
// One wave (32 lanes) per row.
// ---------------------------------------------------------------------------
__global__ void dl_row_invnorm(const float* __restrict__ cb, float* __restrict__ inv) {
    const int row  = blockIdx.x;
    const int lane = threadIdx.x;
    float ss = 0.0f;
    #pragma unroll
    for (int j = lane; j < EE; j += 32) {
        float x = cb[row * EE + j];
        ss = fmaf(x, x, ss);
    }
    #pragma unroll
    for (int off = 16; off > 0; off >>= 1) ss += __shfl_xor(ss, off, 32);
    if (lane == 0) {
        float n = sqrtf(ss);
        inv[row] = 1.0f / fmaxf(n, 1e-12f);
    }
}

// ---------------------------------------------------------------------------
// Gram[m,n] = dot(cb[m,:], cb[n,:]) * inv[m] * inv[n]  via V_WMMA_F32_16X16X4_F32
// One wave per 16x16 tile. blockIdx.y -> m tile, blockIdx.x -> n tile.
//
// ISA fragment layouts (wave32):
//   A (16x4 f32, 2 VGPRs): M = lane&15, K = 2*(lane>>4) + vgpr
//   B ( 4x16 f32, 2 VGPRs): N = lane&15, K = 2*(lane>>4) + vgpr
//   C/D (16x16 f32, 8 VGPRs): M = vgpr + 8*(lane>>4), N = lane&15
// Since B[k,n] = cb[n0+n, k], A and B fragments load with the same mapping.
// ---------------------------------------------------------------------------
__global__ void dl_gram_wmma(const float* __restrict__ cb,
                             const float* __restrict__ inv,
                             float* __restrict__ gram) {
    const int lane  = threadIdx.x;          // 0..31
    const int m0    = blockIdx.y * 16;
    const int n0    = blockIdx.x * 16;
    const int r     = lane & 15;
    const int koff  = (lane >> 4) * 2;      // 0 or 2

    const float* __restrict__ arow = cb + (size_t)(m0 + r) * EE;
    const float* __restrict__ brow = cb + (size_t)(n0 + r) * EE;

    v8f c = {};
    for (int k = 0; k < EE; k += 4) {
        v2f a = *(const v2f*)(arow + k + koff);   // 8B aligned: k+koff is even
        v2f b = *(const v2f*)(brow + k + koff);
        c = __builtin_amdgcn_wmma_f32_16x16x4_f32(
                /*neg_a=*/false, a, /*neg_b=*/false, b,
                /*c_mod=*/(short)0, c, /*reuse_a=*/false, /*reuse_b=*/false);
    }

    const int half = lane >> 4;
    const int col  = lane & 15;
    const float sn = inv[n0 + col];
    #pragma unroll
    for (int v = 0; v < 8; ++v) {
        const int m = m0 + v + 8 * half;
        gram[(size_t)m * DD + n0 + col] = c[v] * inv[m] * sn;
    }
}

// ---------------------------------------------------------------------------
// Fused top-8 + Gram gather/accumulate. One wave per token (grid-stride).
// Lane layout of the 512 values: d = chunk*128 + lane*4 + q  (float4 loads,
// fully coalesced B128). Top-8 extracted by 8 rounds of wave argmax with
// tie-break toward the smaller index (matches lax.top_k descending/stable).
// Each lane accumulates 2 of the 64 (g,h) pairs: p = lane and p = lane+32.
// ---------------------------------------------------------------------------
__global__ void dl_topk_accum(const float* __restrict__ pose,
                              const float* __restrict__ gram,
                              float* __restrict__ acc64) {
    const int lane   = threadIdx.x & 31;
    const int wid    = blockIdx.x * (blockDim.x >> 5) + (threadIdx.x >> 5);
    const int nwaves = gridDim.x * (blockDim.x >> 5);

    // pair p0 = lane -> (g0, h);  pair p1 = lane+32 -> (g0+4, h)
    const int g0 = lane >> 3;
    const int g1 = g0 + 4;
    const int h  = lane & 7;

    float a0 = 0.0f, a1 = 0.0f;

    for (int tok = wid; tok < BN; tok += nwaves) {
        const float4* __restrict__ xp = (const float4*)(pose + (size_t)tok * DD);
        float v[16];
        #pragma unroll
        for (int ch = 0; ch < 4; ++ch) {
            float4 f = xp[ch * 32 + lane];
            v[ch * 4 + 0] = f.x;
            v[ch * 4 + 1] = f.y;
            v[ch * 4 + 2] = f.z;
            v[ch * 4 + 3] = f.w;
        }

        unsigned removed = 0u;
        int ig0 = 0, ig1 = 0, ih = 0;

        #pragma unroll
        for (int t = 0; t < GN; ++t) {
            // lane-local argmax over surviving slots (constant-indexed, no spill)
            float best = -INFINITY;
            int bj = 0;
            #pragma unroll
            for (int j = 0; j < 16; ++j) {
                float val = ((removed >> j) & 1u) ? -INFINITY : v[j];
                if (val > best) { best = val; bj = j; }
            }
            int bd = ((bj >> 2) << 7) + lane * 4 + (bj & 3);   // global d index

            // wave32 argmax, tie -> smaller d
            #pragma unroll
            for (int off = 16; off > 0; off >>= 1) {
                float ov = __shfl_xor(best, off, 32);
                int   od = __shfl_xor(bd,   off, 32);
                if (ov > best || (ov == best && od < bd)) { best = ov; bd = od; }
            }

            // owner lane retires the winning slot
            if (((bd >> 2) & 31) == lane) {
                int j = ((bd >> 7) << 2) | (bd & 3);
                removed |= (1u << j);
            }

            if (t == g0) ig0 = bd;
            if (t == g1) ig1 = bd;
            if (t == h)  ih  = bd;
        }

        a0 += gram[(size_t)ig0 * DD + ih];
        a1 += gram[(size_t)ig1 * DD + ih];
    }

    atomicAdd(&acc64[lane],      a0);
    atomicAdd(&acc64[lane + 32], a1);
}

// ---------------------------------------------------------------------------
// loss = sum_{g,h} | acc[g*8+h]/BN - (g==h) |   (diag <=> index % 9 == 0)
// ---------------------------------------------------------------------------
__global__ void dl_finalize(const float* __restrict__ acc64, float* __restrict__ out) {
    __shared__ float s[64];
    const int t = threadIdx.x;    // 64 threads
    const float m = acc64[t] * (1.0f / (float)BN);
    const float e = ((t % 9) == 0) ? 1.0f : 0.0f;
    s[t] = fabsf(m - e);
    __syncthreads();
    if (t == 0) {
        float sum = 0.0f;
        #pragma unroll
        for (int i = 0; i < 64; ++i) sum += s[i];
        out[0] = sum;
    }
}

// ---------------------------------------------------------------------------
// kernel_launch: inputs are [pose_code (B*N*D f32), codebook (D*E f32)],
// output is a single f32 scalar. Workspace layout:
//   [0, 512)              inv row norms
//   [512, 512+512*512)    Gram matrix (1 MB)
//   then 64 floats        pair accumulator
// ---------------------------------------------------------------------------
extern "C" void kernel_launch(void* const* d_in, const int* in_sizes, int n_in,
                              void* d_out, int out_size, void* d_ws, size_t ws_size,
                              hipStream_t stream) {
    (void)in_sizes; (void)n_in; (void)out_size; (void)ws_size;

    const float* pose = (const float*)d_in[0];   // [B,N,D] f32
    const float* cb   = (const float*)d_in[1];   // [D,E]  f32
    float*       out  = (float*)d_out;

    float* ws_inv  = (float*)d_ws;               // 512
    float* ws_gram = ws_inv + DD;                // 512*512
    float* ws_acc  = ws_gram + (size_t)DD * DD;  // 64

    dl_zero_acc<<<1, 64, 0, stream>>>(ws_acc);

    dl_row_invnorm<<<DD, 32, 0, stream>>>(cb, ws_inv);

    dim3 ggrid(DD / 16, DD / 16);                // 32 x 32 tiles
    dl_gram_wmma<<<ggrid, 32, 0, stream>>>(cb, ws_inv, ws_gram);

    dl_topk_accum<<<1024, 256, 0, stream>>>(pose, ws_gram, ws_acc);

    dl_finalize<<<1, 64, 0, stream>>>(ws_acc, out);
}